// IntDVF_72481868087625
// MI455X (gfx1250) — compile-verified
//
#include <hip/hip_runtime.h>

// Scaling-and-squaring DVF integration (DeepReg-style), B=2, 128^3, C=3, f32.
//   ddf0 = dvf * 2^-7
//   repeat 7x: ddf_next[p] = ddf[p] + trilerp(ddf, clamp(p + ddf[p]))
// Ping-pong: init -> ws; steps alternate ws->out, out->ws, ... step7 -> out.
// Design: gather-bound, whole working set (2 x 50.3MB) is L2-resident on the
// MI455X's 192MB L2. Per voxel: 9 loads (8x b96 corner gathers + own b96),
// 1x b96 store. Channel math packed as float2+float so the backend emits
// v_pk_fma_f32 / v_pk_mul_f32 (VOP3P packed f32, 2x rate) for 2 of 3 channels.

#define DIM   128
#define DMASK 127
#define CH    3

typedef float f2 __attribute__((ext_vector_type(2)));

struct V3 { f2 xy; float z; };

__device__ __forceinline__ V3 ld3(const float* __restrict__ p) {
    V3 r;
    r.xy = (f2){p[0], p[1]};   // offsets 0,4,8 from one base -> global_load_b96
    r.z  = p[2];
    return r;
}

// a + w*(b-a); -ffp-contract=fast turns the vector form into v_pk_fma_f32
__device__ __forceinline__ f2 lerp2(f2 a, f2 b, f2 w)          { return a + w * (b - a); }
__device__ __forceinline__ float lerp1(float a, float b, float w) { return fmaf(w, b - a, a); }

__global__ __launch_bounds__(256) void ddf_init(const float4* __restrict__ dvf,
                                                float4* __restrict__ cur, int n4) {
    int t = blockIdx.x * 256 + threadIdx.x;
    if (t < n4) {
        float4 v = dvf[t];
        const float s = 0.0078125f;  // 2^-7
        v.x *= s; v.y *= s; v.z *= s; v.w *= s;   // -> v_pk_mul_f32 pair
        cur[t] = v;
    }
}

__global__ __launch_bounds__(256) void ddf_step(const float* __restrict__ cur,
                                                float* __restrict__ nxt, int nvox) {
    int t = blockIdx.x * 256 + threadIdx.x;
    if (t >= nvox) return;

    // t = ((b*128 + z)*128 + y)*128 + x  (x fastest -> coalesced lanes)
    int x = t & DMASK;
    int y = (t >> 7)  & DMASK;
    int z = (t >> 14) & DMASK;
    int b = t >> 21;

    const float* __restrict__ base = cur + (size_t)b * (size_t)(DIM * DIM * DIM * CH);
    V3 d = ld3(cur + (size_t)t * CH);

    // coords (dim order: 0->D(z) from ch0, 1->H(y) from ch1, 2->W(x) from ch2)
    float c0 = fminf(fmaxf((float)z + d.xy.x, 0.0f), 127.0f);
    float c1 = fminf(fmaxf((float)y + d.xy.y, 0.0f), 127.0f);
    float c2 = fminf(fmaxf((float)x + d.z,    0.0f), 127.0f);

    float f0 = floorf(c0), f1 = floorf(c1), f2c = floorf(c2);
    int   i0 = (int)f0,    j0 = (int)f1,    k0  = (int)f2c;
    int   i1 = min(i0 + 1, DMASK);
    int   j1 = min(j0 + 1, DMASK);
    int   k1 = min(k0 + 1, DMASK);
    float w0 = c0 - f0, w1 = c1 - f1, w2 = c2 - f2c;

    f2 w0v = (f2){w0, w0};
    f2 w1v = (f2){w1, w1};
    f2 w2v = (f2){w2, w2};

    int r00 = ((i0 << 7) + j0) << 7;  // row bases (voxel units)
    int r01 = ((i0 << 7) + j1) << 7;
    int r10 = ((i1 << 7) + j0) << 7;
    int r11 = ((i1 << 7) + j1) << 7;

    V3 v000 = ld3(base + (size_t)(r00 + k0) * CH);
    V3 v001 = ld3(base + (size_t)(r00 + k1) * CH);
    V3 v010 = ld3(base + (size_t)(r01 + k0) * CH);
    V3 v011 = ld3(base + (size_t)(r01 + k1) * CH);
    V3 v100 = ld3(base + (size_t)(r10 + k0) * CH);
    V3 v101 = ld3(base + (size_t)(r10 + k1) * CH);
    V3 v110 = ld3(base + (size_t)(r11 + k0) * CH);
    V3 v111 = ld3(base + (size_t)(r11 + k1) * CH);

    // lerp along k (w2), then j (w1), then i (w0); ch{0,1} packed, ch2 scalar
    f2    a00 = lerp2(v000.xy, v001.xy, w2v);  float a00z = lerp1(v000.z, v001.z, w2);
    f2    a01 = lerp2(v010.xy, v011.xy, w2v);  float a01z = lerp1(v010.z, v011.z, w2);
    f2    a10 = lerp2(v100.xy, v101.xy, w2v);  float a10z = lerp1(v100.z, v101.z, w2);
    f2    a11 = lerp2(v110.xy, v111.xy, w2v);  float a11z = lerp1(v110.z, v111.z, w2);

    f2    b0  = lerp2(a00, a01, w1v);          float b0z  = lerp1(a00z, a01z, w1);
    f2    b1  = lerp2(a10, a11, w1v);          float b1z  = lerp1(a10z, a11z, w1);

    f2    s   = lerp2(b0, b1, w0v);            float sz   = lerp1(b0z, b1z, w0);

    f2    oxy = d.xy + s;                      // v_pk_add_f32
    float oz  = d.z + sz;

    float* __restrict__ o = nxt + (size_t)t * CH;
    o[0] = oxy.x;    // offsets 0,4,8 -> global_store_b96
    o[1] = oxy.y;
    o[2] = oz;
}

extern "C" void kernel_launch(void* const* d_in, const int* in_sizes, int n_in,
                              void* d_out, int out_size, void* d_ws, size_t ws_size,
                              hipStream_t stream) {
    (void)n_in; (void)out_size; (void)ws_size;
    const float* dvf = (const float*)d_in[0];
    float* out = (float*)d_out;
    float* ws  = (float*)d_ws;

    const int n    = in_sizes[0];     // 2*128^3*3 = 12,582,912 floats
    const int nvox = n / CH;          // 4,194,304 voxels
    const int n4   = n / 4;           // exactly divisible

    // ddf0 = dvf * 2^-7  -> ws (L2-resident from here on: 2x50.3MB << 192MB L2)
    ddf_init<<<(n4 + 255) / 256, 256, 0, stream>>>((const float4*)dvf, (float4*)ws, n4);

    // 7 squaring steps, ping-pong ws <-> out; final step (s=6, even) lands in out.
    const float* src = ws;
    for (int s = 0; s < 7; ++s) {
        float* dst = (s & 1) ? ws : out;
        ddf_step<<<(nvox + 255) / 256, 256, 0, stream>>>(src, dst, nvox);
        src = dst;
    }
}